// Rotation_agg_20813411516915
// MI455X (gfx1250) — compile-verified
//
#include <hip/hip_runtime.h>
#include <stdint.h>

// ---------------------------------------------------------------------------
// Rotation-aggregate (RotatE-style metapath aggregation), MI455X / gfx1250.
//
// Memory-bound: 1.28 GB traffic -> ~55us floor @ 23.3 TB/s. Strategy:
//   * TDM (tensor_load_to_lds) DMA streams 16-node (16KB) tiles HBM -> LDS,
//     double buffered on TENSORcnt; zero VALU cost for the load path.
//   * 8 wave32s rotate + average out of LDS (bank-conflict-free ds_load_b64),
//     write coalesced global_store_b64.
//   * Rotation table F[l][d] (3x32 unit complex) computed once per block.
// ---------------------------------------------------------------------------

typedef unsigned int u32x4 __attribute__((ext_vector_type(4)));
typedef int          i32x8 __attribute__((ext_vector_type(8)));
typedef int          i32x4 __attribute__((ext_vector_type(4)));

#define NPT 16                      // nodes per tile
#define TILE_FLOATS (NPT * 256)     // 4096 floats = 16 KB per tile buffer

// Issue a TDM load of one tile (NPT nodes x 256 floats, contiguous) into LDS.
// D# built per CDNA5 ISA ch. 8 (CDNA5 ISA): 2D tile, data_size=4B.
// This toolchain's builtin is the 6-arg form (clang-23 / therock-10.0).
__device__ __forceinline__ void tdm_load_tile(const float* gbase, uint32_t lds_off)
{
    uint64_t ga = (uint64_t)(uintptr_t)gbase;

    u32x4 g0;
    g0.x = 1u;                                            // count=1 (valid), user mode
    g0.y = lds_off;                                       // lds_addr (bytes)
    g0.z = (uint32_t)(ga & 0xFFFFFFFFu);                  // global_addr[31:0]
    g0.w = (uint32_t)((ga >> 32) & 0x1FFFFFFu)            // global_addr[56:32]
         | 0x80000000u;                                   // type=2 ("image")

    i32x8 g1;
    g1[0] = (int)(2u << 16);                              // data_size=2 -> 4 bytes
    g1[1] = (int)(256u << 16);                            // tensor_dim0 = 256 (row = one node)
    g1[2] = (int)((uint32_t)NPT << 16);                   // tensor_dim1 = NPT rows
    g1[3] = (int)(256u << 16);                            // tile_dim0 = 256
    g1[4] = NPT;                                          // tile_dim1 = NPT, tile_dim2 = 0
    g1[5] = 256;                                          // tensor_dim0_stride = 256 elements
    g1[6] = 0;                                            // stride0 hi, stride1 lo
    g1[7] = 0;                                            // stride1 hi

    i32x4 g2;                                             // tensor_dim2=1, tensor_dim3=1
    g2[0] = 1; g2[1] = 1; g2[2] = 0; g2[3] = 0;           // tile_dim3 = 0 (unused)
    i32x4 g3;                                             // tensor_dim4=1, tile_dim4=0
    g3[0] = 0; g3[1] = (int)(1u << 16); g3[2] = 0; g3[3] = 0;

    i32x8 gx;                                             // extra operand (6-arg form);
    gx[0] = 0; gx[1] = 0; gx[2] = 0; gx[3] = 0;           // zero-filled
    gx[4] = 0; gx[5] = 0; gx[6] = 0; gx[7] = 0;

    __builtin_amdgcn_tensor_load_to_lds(g0, g1, g2, g3, gx, 0);
}

__global__ __launch_bounds__(256)
void rot_agg_kernel(const float* __restrict__ feat,
                    const float* __restrict__ rvec,
                    float* __restrict__ out,
                    int numTiles, long N)
{
    __shared__ float sTile[2][TILE_FLOATS];   // 32 KB double buffer
    __shared__ float sF[3 * 64];              // F[l][d] as interleaved complex

    const int tid = threadIdx.x;

    // --- Build rotation table F (lanes 0..31, one complex index d each) -----
    // ETYPES=[0,3,5] over rv_full=[rv0,c0,rv1,c1,rv2,c2,rv3,c3]:
    //   F3 = 1;  F2 = conj(rv2);  F1 = F2*conj(rv1);  F0 = F1*rv0
    if (tid < 32) {
        int d = tid;
        float a0x = rvec[      2*d], a0y = rvec[      2*d + 1];
        float a1x = rvec[ 64 + 2*d], a1y = rvec[ 64 + 2*d + 1];
        float a2x = rvec[128 + 2*d], a2y = rvec[128 + 2*d + 1];
        float i0 = 1.0f / fmaxf(sqrtf(a0x*a0x + a0y*a0y), 1e-12f);
        float i1 = 1.0f / fmaxf(sqrtf(a1x*a1x + a1y*a1y), 1e-12f);
        float i2 = 1.0f / fmaxf(sqrtf(a2x*a2x + a2y*a2y), 1e-12f);
        float r0x = a0x*i0, r0y =  a0y*i0;
        float r1x = a1x*i1, r1y = -a1y*i1;    // conj
        float r2x = a2x*i2, r2y = -a2y*i2;    // conj
        float F2x = r2x,               F2y = r2y;
        float F1x = F2x*r1x - F2y*r1y, F1y = F2x*r1y + F2y*r1x;
        float F0x = F1x*r0x - F1y*r0y, F0y = F1x*r0y + F1y*r0x;
        sF[      2*d] = F0x; sF[      2*d + 1] = F0y;
        sF[ 64 + 2*d] = F1x; sF[ 64 + 2*d + 1] = F1y;
        sF[128 + 2*d] = F2x; sF[128 + 2*d + 1] = F2y;
    }
    __syncthreads();

    const int d  = tid & 31;     // complex index: lane sweep -> 64 distinct LDS banks
    const int ng = tid >> 5;     // node sub-group 0..7
    const float F0x = sF[      2*d], F0y = sF[      2*d + 1];
    const float F1x = sF[ 64 + 2*d], F1y = sF[ 64 + 2*d + 1];
    const float F2x = sF[128 + 2*d], F2y = sF[128 + 2*d + 1];

    // --- TDM double-buffered tile pipeline ---------------------------------
    const bool issuer = (tid < 32);          // wave 0 only (TDM issues per wave)
    const long t0     = blockIdx.x;
    const long stride = gridDim.x;

    if (issuer && t0 < numTiles)
        tdm_load_tile(feat + t0 * (long)TILE_FLOATS,
                      (uint32_t)(uintptr_t)&sTile[0][0]);

    int buf = 0;
    for (long t = t0; t < numTiles; t += stride) {
        long tn = t + stride;
        if (issuer) {
            if (tn < numTiles) {
                tdm_load_tile(feat + tn * (long)TILE_FLOATS,
                              (uint32_t)(uintptr_t)&sTile[buf ^ 1][0]);
                __builtin_amdgcn_s_wait_tensorcnt(1);  // tile t complete
            } else {
                __builtin_amdgcn_s_wait_tensorcnt(0);
            }
        }
        __syncthreads();                                // LDS tile visible to all waves

        const float* sb = &sTile[buf][0];
        #pragma unroll
        for (int k = 0; k < 2; ++k) {
            int node = ng + k * 8;
            const float* p = sb + node * 256 + 2 * d;
            float2 e0 = *(const float2*)(p);
            float2 e1 = *(const float2*)(p + 64);
            float2 e2 = *(const float2*)(p + 128);
            float2 e3 = *(const float2*)(p + 192);
            float ax = e3.x + e0.x*F0x - e0.y*F0y
                            + e1.x*F1x - e1.y*F1y
                            + e2.x*F2x - e2.y*F2y;
            float ay = e3.y + e0.x*F0y + e0.y*F0x
                            + e1.x*F1y + e1.y*F1x
                            + e2.x*F2y + e2.y*F2x;
            float2 o; o.x = ax * 0.25f; o.y = ay * 0.25f;
            *(float2*)(out + (t * NPT + node) * 64 + 2 * d) = o;   // coalesced b64
        }
        __syncthreads();                 // tile buf free before it is re-filled
        buf ^= 1;
    }

    // --- Tail nodes (N % NPT; zero for N=1e6) via direct global loads ------
    long tailStart = (long)numTiles * NPT;
    long nrem = N - tailStart;
    if (nrem > 0) {
        for (long idx = (long)blockIdx.x * 256 + tid; idx < nrem * 32;
             idx += stride * 256) {
            long n = tailStart + (idx >> 5);           // idx%32 == d (stride%32==0)
            const float* p = feat + n * 256 + 2 * d;
            float2 e0 = *(const float2*)(p);
            float2 e1 = *(const float2*)(p + 64);
            float2 e2 = *(const float2*)(p + 128);
            float2 e3 = *(const float2*)(p + 192);
            float ax = e3.x + e0.x*F0x - e0.y*F0y + e1.x*F1x - e1.y*F1y
                            + e2.x*F2x - e2.y*F2y;
            float ay = e3.y + e0.x*F0y + e0.y*F0x + e1.x*F1y + e1.y*F1x
                            + e2.x*F2y + e2.y*F2x;
            float2 o; o.x = ax * 0.25f; o.y = ay * 0.25f;
            *(float2*)(out + n * 64 + 2 * d) = o;
        }
    }
}

extern "C" void kernel_launch(void* const* d_in, const int* in_sizes, int n_in,
                              void* d_out, int out_size, void* d_ws, size_t ws_size,
                              hipStream_t stream)
{
    const float* feat = (const float*)d_in[0];   // (N, 4, 64) f32
    const float* rvec = (const float*)d_in[1];   // (4, 32, 2) f32
    float* out = (float*)d_out;                  // (N, 64) f32

    long N = (long)in_sizes[0] / 256;            // 1,000,000
    int numTiles = (int)(N / NPT);               // 62,500 full tiles
    int grid = numTiles < 2048 ? (numTiles > 0 ? numTiles : 1) : 2048;

    hipLaunchKernelGGL(rot_agg_kernel, dim3(grid), dim3(256), 0, stream,
                       feat, rvec, out, numTiles, N);
}